// Cell_64063732187495
// MI455X (gfx1250) — compile-verified
//
#include <hip/hip_runtime.h>
#include <hip/hip_bf16.h>

#define N_NODES 50000
#define N_EDGES 600000
#define CH 128
#define NC (N_NODES * CH)   // 6,400,000 floats per [N,C] buffer
#define LSTRIDE 34          // LDS row stride (floats): 8B-aligned rows, conflict-free reads

typedef __attribute__((ext_vector_type(2))) float v2f;
typedef __attribute__((ext_vector_type(8))) float v8f;

// ---------------------------------------------------------------------------
// small helpers
// ---------------------------------------------------------------------------
__device__ __forceinline__ void atomicMaxF(float* addr, float v) {
  // sign-split trick: int ordering matches float for >=0, reversed (as uint) for <0
  if (v >= 0.0f) atomicMax((int*)addr, __float_as_int(v));
  else           atomicMin((unsigned int*)addr, __float_as_uint(v));
}

__global__ void fill_kernel(float* p, float v, int n) {
  int i = blockIdx.x * 256 + threadIdx.x;
  if (i < n) p[i] = v;
}

__global__ void deg_kernel(const int* __restrict__ dst, float* __restrict__ deg, int e) {
  int i = blockIdx.x * 256 + threadIdx.x;
  if (i < e) atomicAdd(&deg[dst[i]], 1.0f);
}

// per-column sum / sumsq over [N,128] (for BatchNorm)
__global__ void colstats_kernel(const float* __restrict__ h, float* __restrict__ stats, int n) {
  int c = threadIdx.x & (CH - 1);
  int r = blockIdx.x * 2 + (threadIdx.x >> 7);
  float s = 0.f, s2 = 0.f;
  for (; r < n; r += gridDim.x * 2) {
    float v = h[(size_t)r * CH + c];
    s += v; s2 += v * v;
  }
  atomicAdd(&stats[c], s);
  atomicAdd(&stats[CH + c], s2);
}

__global__ void bn_relu_kernel(const float* __restrict__ h, const float* __restrict__ stats,
                               const float* __restrict__ gamma, const float* __restrict__ beta,
                               float* __restrict__ out) {
  int i = blockIdx.x * 256 + threadIdx.x;
  if (i >= NC) return;
  int c = i & (CH - 1);
  float mu  = stats[c] * (1.0f / N_NODES);
  float var = stats[CH + c] * (1.0f / N_NODES) - mu * mu;
  float v = (h[i] - mu) * rsqrtf(var + 1e-5f) * gamma[c] + beta[c];
  out[i] = v > 0.f ? v : 0.f;
}

// ---------------------------------------------------------------------------
// edge aggregation: 32 lanes per edge, float4 per lane (coalesced gather)
// ---------------------------------------------------------------------------
__global__ void agg_kernel(const float* __restrict__ h, int hstride,
                           const int* __restrict__ src, const int* __restrict__ dst,
                           float* __restrict__ asum, float* __restrict__ amax) {
  int tid = blockIdx.x * 256 + threadIdx.x;
  if (tid >= N_EDGES * 32) return;
  int e = tid >> 5, q = tid & 31;
  int s = src[e], d = dst[e];
  float4 v = *(const float4*)(h + (size_t)s * hstride + q * 4);
  float* sb = asum + (size_t)d * CH + q * 4;
  float* mb = amax + (size_t)d * CH + q * 4;
  atomicAdd(sb + 0, v.x); atomicAdd(sb + 1, v.y);
  atomicAdd(sb + 2, v.z); atomicAdd(sb + 3, v.w);
  atomicMaxF(mb + 0, v.x); atomicMaxF(mb + 1, v.y);
  atomicMaxF(mb + 2, v.z); atomicMaxF(mb + 3, v.w);
}

// sum -> mean (deg clamp), max: -inf (no in-edges) -> 0
__global__ void finalize_agg_kernel(float* __restrict__ mean, float* __restrict__ amax,
                                    const float* __restrict__ deg) {
  int i = blockIdx.x * 256 + threadIdx.x;
  if (i >= NC) return;
  float d = deg[i >> 7];
  if (d < 1.f) d = 1.f;
  mean[i] = mean[i] / d;
  float m = amax[i];
  amax[i] = __builtin_isfinite(m) ? m : 0.f;
}

// ---------------------------------------------------------------------------
// build stacked/folded weights for step t:
//   block j (< cnt):  w2[m]*Wg[m] + 0.9*w4[m]*Wii[m],  m = off+j
//   block cnt (x0):   0.1 * sum_j w4[m]*Wii[m]
// ---------------------------------------------------------------------------
__global__ void wstack_kernel(const float* __restrict__ Wg, const float* __restrict__ Wii,
                              const float* __restrict__ wts, float* __restrict__ wstk,
                              int off, int cnt) {
  int idx = blockIdx.x * 256 + threadIdx.x;
  int total = (cnt + 1) * (CH * CH);
  if (idx >= total) return;
  int blk = idx / (CH * CH);
  int el  = idx - blk * (CH * CH);
  if (blk < cnt) {
    int m = off + blk;
    wstk[idx] = wts[m * 5 + 2] * Wg[(size_t)m * CH * CH + el]
              + 0.9f * wts[m * 5 + 4] * Wii[(size_t)m * CH * CH + el];
  } else {
    float acc = 0.f;
    for (int j = 0; j < cnt; j++) {
      int m = off + j;
      acc += wts[m * 5 + 4] * Wii[(size_t)m * CH * CH + el];
    }
    wstk[idx] = 0.1f * acc;
  }
}

// ---------------------------------------------------------------------------
// elementwise part of a step: OUT_t = sum_j ( w1*state_j + w3*aggmax_j )
// ---------------------------------------------------------------------------
struct EwArgs {
  const float* st[5];
  int          sstride[5];
  const float* amax[5];
  float*       out;      // d_out + t*128, row stride 512
  const float* wts;
  int off, cnt;
};

__global__ void ew_kernel(EwArgs a) {
  int i = blockIdx.x * 256 + threadIdx.x;
  if (i >= NC) return;
  int r = i >> 7, c = i & (CH - 1);
  float acc = 0.f;
  for (int j = 0; j < a.cnt; j++) {
    float w1 = a.wts[(a.off + j) * 5 + 1];
    float w3 = a.wts[(a.off + j) * 5 + 3];
    acc += w1 * a.st[j][(size_t)r * a.sstride[j] + c] + w3 * a.amax[j][i];
  }
  a.out[(size_t)r * 512 + c] = acc;
}

// ---------------------------------------------------------------------------
// WMMA fp32 GEMM: OUT[N,128] (+)= [A_0 | A_1 | ... | A_{nblk-1}] @ B
//   each A block is [N,128] row-stride 128; B is [nblk*128, 128] row-major
//   block = 256 threads = 8 waves; wave w owns the 16x16 tile at cols w*16
//   A staged via double-buffered GLOBAL_LOAD_ASYNC_TO_LDS_B64 (ASYNCcnt):
//   chunk i+1 streams into the alternate buffer while the 8 WMMAs of chunk i run
// ---------------------------------------------------------------------------
struct GemmArgs {
  const float* a[6];
  const float* b;
  float*       out;
  int ostride;
  int nblk;
  int accumulate;
};

__global__ __launch_bounds__(256) void wmma_gemm_kernel(GemmArgs g) {
  __shared__ float lds[2][16 * LSTRIDE];
  const int lane = threadIdx.x & 31;
  const int wave = threadIdx.x >> 5;       // col tile 0..7
  const int row0 = blockIdx.x * 16;
  const int col0 = wave * 16;
  const int hl   = lane >> 4;              // half-wave (K split for A/B frags)
  const int l    = lane & 15;

  // async staging map: thread -> one 8B segment of the 16x32 A chunk
  const int m_ld = threadIdx.x >> 4;       // row 0..15
  const int seg  = threadIdx.x & 15;       // 2-float segment 0..15
  // flat LDS aperture keeps the byte offset in addr[31:0]
  const unsigned lb0 = (unsigned)(unsigned long long)&lds[0][0];
  const unsigned lb1 = (unsigned)(unsigned long long)&lds[1][0];
  const unsigned lrow = (unsigned)(m_ld * LSTRIDE + seg * 2) * 4u;

  v8f acc = {};
  if (g.accumulate) {
#pragma unroll
    for (int i = 0; i < 8; i++)
      acc[i] = g.out[(size_t)(row0 + i + hl * 8) * g.ostride + col0 + l];
  }

  const int nchunks = g.nblk * 4;          // 32-wide K chunks
  {                                        // prologue: chunk 0 -> buf0
    const float* gp = g.a[0] + (size_t)(row0 + m_ld) * CH + seg * 2;
    unsigned loff = lb0 + lrow;
    asm volatile("global_load_async_to_lds_b64 %0, %1, off"
                 :: "v"(loff), "v"(gp) : "memory");
  }

  for (int c = 0; c < nchunks; c++) {
    if (c + 1 < nchunks) {                 // issue next chunk into alt buffer
      int blk = (c + 1) >> 2, kc = (c + 1) & 3;
      const float* gp = g.a[blk] + (size_t)(row0 + m_ld) * CH + kc * 32 + seg * 2;
      unsigned loff = (((c + 1) & 1) ? lb1 : lb0) + lrow;
      asm volatile("global_load_async_to_lds_b64 %0, %1, off"
                   :: "v"(loff), "v"(gp) : "memory");
      asm volatile("s_wait_asynccnt 0x1" ::: "memory");   // chunk c done, c+1 in flight
    } else {
      asm volatile("s_wait_asynccnt 0x0" ::: "memory");
    }
    __syncthreads();                       // chunk c visible to all waves

    const float* lbuf = lds[c & 1];
    const float* Bp = g.b + (size_t)((c >> 2) * CH + (c & 3) * 32) * CH + col0 + l;
#pragma unroll
    for (int kk = 0; kk < 32; kk += 4) {
      v2f af, bf;
      // A 16x4 frag: lanes0-15 hold K{0,1}, lanes16-31 hold K{2,3} (M = l)
      af.x = lbuf[l * LSTRIDE + kk + hl * 2];
      af.y = lbuf[l * LSTRIDE + kk + hl * 2 + 1];
      // B 4x16 frag: rows striped across lanes (N = l), K split like A
      bf.x = Bp[(size_t)(kk + hl * 2) * CH];
      bf.y = Bp[(size_t)(kk + hl * 2 + 1) * CH];
      acc = __builtin_amdgcn_wmma_f32_16x16x4_f32(
          false, af, false, bf, (short)0, acc, false, false);
    }
    __syncthreads();                       // all waves done reading buf before reuse
  }

#pragma unroll
  for (int i = 0; i < 8; i++)
    g.out[(size_t)(row0 + i + hl * 8) * g.ostride + col0 + l] = acc[i];
}

// ---------------------------------------------------------------------------
// host orchestration
// ---------------------------------------------------------------------------
extern "C" void kernel_launch(void* const* d_in, const int* in_sizes, int n_in,
                              void* d_out, int out_size, void* d_ws, size_t ws_size,
                              hipStream_t stream) {
  const float* s0   = (const float*)d_in[0];
  const float* s1   = (const float*)d_in[1];
  const float* x0   = (const float*)d_in[2];
  const float* W0   = (const float*)d_in[3];
  const float* g0   = (const float*)d_in[4];
  const float* b0   = (const float*)d_in[5];
  const float* W1   = (const float*)d_in[6];
  const float* g1   = (const float*)d_in[7];
  const float* b1   = (const float*)d_in[8];
  const float* Wg   = (const float*)d_in[9];
  const float* Wii  = (const float*)d_in[10];
  const float* wts  = (const float*)d_in[11];
  const int*   ei   = (const int*)d_in[12];
  const int*   src  = ei;
  const int*   dst  = ei + N_EDGES;
  float*       out  = (float*)d_out;

  // workspace partition
  float* f = (float*)d_ws;
  float* deg   = f;  f += N_NODES;
  float* stats = f;  f += 256;
  float* st0   = f;  f += NC;
  float* st1   = f;  f += NC;
  float* amean[5]; for (int j = 0; j < 5; j++) { amean[j] = f; f += NC; }
  float* amax [5]; for (int j = 0; j < 5; j++) { amax [j] = f; f += NC; }
  float* wstk  = f;  f += 6 * CH * CH;
  float* tmp   = amean[0];   // preprocess scratch; aggregation starts after preprocess

  const int GB  = (NC + 255) / 256;        // [N,C] elementwise grid
  const int GE  = (N_EDGES * 32 + 255) / 256;
  const int GG  = N_NODES / 16;            // 3125 (exact)
  const int off[4] = {0, 2, 5, 9};

  // degree
  hipMemsetAsync(deg, 0, N_NODES * sizeof(float), stream);
  deg_kernel<<<(N_EDGES + 255) / 256, 256, 0, stream>>>(dst, deg, N_EDGES);

  // preprocess: state = relu(batchnorm(s @ W))
  auto preprocess = [&](const float* s, const float* W, const float* gam,
                        const float* bet, float* stout) {
    GemmArgs ga = {};
    ga.a[0] = s; ga.b = W; ga.out = tmp; ga.ostride = CH; ga.nblk = 1; ga.accumulate = 0;
    wmma_gemm_kernel<<<GG, 256, 0, stream>>>(ga);
    hipMemsetAsync(stats, 0, 256 * sizeof(float), stream);
    colstats_kernel<<<512, 256, 0, stream>>>(tmp, stats, N_NODES);
    bn_relu_kernel<<<GB, 256, 0, stream>>>(tmp, stats, gam, bet, stout);
  };
  preprocess(s0, W0, g0, b0, st0);
  preprocess(s1, W1, g1, b1, st1);

  // state pointer/stride for j (states 2..5 live in d_out, stride 512)
  auto state_ptr    = [&](int j) -> const float* { return j == 0 ? st0 : j == 1 ? st1 : out + (j - 2) * CH; };
  auto state_stride = [&](int j) -> int          { return j < 2 ? CH : 512; };

  // aggregate state j once (dedup: reference re-aggregates 14x, only 5 unique)
  auto aggregate = [&](int j) {
    hipMemsetAsync(amean[j], 0, (size_t)NC * sizeof(float), stream);
    fill_kernel<<<GB, 256, 0, stream>>>(amax[j], -__builtin_inff(), NC);
    agg_kernel<<<GE, 256, 0, stream>>>(state_ptr(j), state_stride(j), src, dst, amean[j], amax[j]);
    finalize_agg_kernel<<<GB, 256, 0, stream>>>(amean[j], amax[j], deg);
  };
  aggregate(0);
  aggregate(1);

  for (int t = 0; t < 4; t++) {
    const int cnt = t + 2;
    // folded/stacked weights: cnt blocks of (w2*Wg + 0.9*w4*Wii) + one x0 block
    wstack_kernel<<<((cnt + 1) * CH * CH + 255) / 256, 256, 0, stream>>>(
        Wg, Wii, wts, wstk, off[t], cnt);

    // elementwise: OUT_t = sum_j (w1*state_j + w3*aggmax_j)
    EwArgs ea = {};
    for (int j = 0; j < cnt; j++) {
      ea.st[j] = state_ptr(j); ea.sstride[j] = state_stride(j); ea.amax[j] = amax[j];
    }
    ea.out = out + t * CH; ea.wts = wts; ea.off = off[t]; ea.cnt = cnt;
    ew_kernel<<<GB, 256, 0, stream>>>(ea);

    // OUT_t += [amean_0 | ... | amean_{cnt-1} | x0] @ wstack
    GemmArgs ga = {};
    for (int j = 0; j < cnt; j++) ga.a[j] = amean[j];
    ga.a[cnt] = x0;
    ga.b = wstk; ga.out = out + t * CH; ga.ostride = 512;
    ga.nblk = cnt + 1; ga.accumulate = 1;
    wmma_gemm_kernel<<<GG, 256, 0, stream>>>(ga);

    if (t < 3) aggregate(t + 2);   // state t+2 now final -> aggregate for later steps
  }
}